// LogSumExpWirelength_33767032881791
// MI455X (gfx1250) — compile-verified
//
#include <hip/hip_runtime.h>

// LogSumExp wirelength (DREAMPlace-style), MI455X / gfx1250, wave32.
//
// Memory-bound segmented reduction: ~128 MB touched => ~5.5 us floor at 23.3 TB/s.
// - one net per thread (deg==5 here; register path for deg<=8)
// - flat_netpin / netpin_start are single-use streams -> non-temporal loads
//   (TH=NT) so the 80 MB `pos` array stays resident in the 192 MB L2 for the
//   random gathers (which keep default RT caching).
// - stable LSE: gamma*(lse(x/g)+lse(-x/g)) =
//     (max-min) + g*(log sum e^{(x-max)/g} + log sum e^{(min-x)/g})
// - wave32 reduction via V_WMMA_F32_16X16X4_F32 with an all-ones B
//   (all-ones is layout-invariant; only documented A/D layouts are relied on),
//   then one shfl_xor(16). Deterministic: block partials -> d_ws, fixed-shape
//   single-block final reduce, no float atomics.

typedef __attribute__((ext_vector_type(2))) float v2f;
typedef __attribute__((ext_vector_type(8))) float v8f;

#define WL_BLOCK 256
#define WL_MAX_BLOCKS 4096

// Sum of one float per lane across all 32 lanes of a wave.
// Requires EXEC = all ones (call from fully-converged code only).
__device__ __forceinline__ float wave_sum32_wmma(float p) {
  v2f a; a.x = p;    a.y = 0.0f;   // A[m][0]=p_m, A[m][2]=p_{m+16}, A[m][1]=A[m][3]=0
  v2f b; b.x = 1.0f; b.y = 1.0f;   // all-ones B: D[m][n] = rowsum_m = p_m + p_{m+16}
  v8f c = {};
  v8f d = __builtin_amdgcn_wmma_f32_16x16x4_f32(
      /*neg_a=*/false, a, /*neg_b=*/false, b,
      /*c_mod=*/(short)0, c, /*reuse_a=*/false, /*reuse_b=*/false);
  // lane<16: sum of D VGPRs = rows 0..7 ; lane>=16: rows 8..15
  float s = d.s0 + d.s1 + d.s2 + d.s3 + d.s4 + d.s5 + d.s6 + d.s7;
  s += __shfl_xor(s, 16, 32);      // total over all 32 lanes, in every lane
  return s;
}

__device__ __forceinline__ float block_sum(float v) {
  __shared__ float smem[WL_BLOCK / 32];
  float ws = wave_sum32_wmma(v);
  const int lane = threadIdx.x & 31;
  const int wid  = threadIdx.x >> 5;
  if (lane == 0) smem[wid] = ws;
  __syncthreads();
  float b = 0.0f;
  if (threadIdx.x == 0) {
#pragma unroll
    for (int i = 0; i < WL_BLOCK / 32; ++i) b += smem[i];
  }
  __syncthreads();
  return b;  // valid in thread 0
}

__global__ void __launch_bounds__(WL_BLOCK)
wl_net_kernel(const float* __restrict__ pos,
              const int* __restrict__ flat_netpin,
              const int* __restrict__ netpin_start,
              float* __restrict__ partials,
              int num_nets, int num_pins) {
  const float g     = 4.0f;        // GAMMA
  const float inv_g = 0.25f;       // 1/GAMMA
  const int stride  = gridDim.x * blockDim.x;

  float acc = 0.0f;
  for (int net = blockIdx.x * blockDim.x + threadIdx.x; net < num_nets; net += stride) {
    // single-use stream: non-temporal (don't pollute L2 holding `pos`)
    const int s   = __builtin_nontemporal_load(&netpin_start[net]);
    const int e   = __builtin_nontemporal_load(&netpin_start[net + 1]);
    const int deg = e - s;
    if (deg <= 0) continue;
    if (deg >= num_pins) continue;   // IGNORE_NET_DEGREE == num_pins in reference

    float mxx = -3.402823e38f, mnx = 3.402823e38f;
    float mxy = -3.402823e38f, mny = 3.402823e38f;
    float spx = 0.0f, snx = 0.0f, spy = 0.0f, sny = 0.0f;

    if (deg <= 8) {
      // register-resident path (this workload: deg == 5)
      float xs[8], ys[8];
#pragma unroll
      for (int i = 0; i < 8; ++i) {
        if (i < deg) {
          const int idx = __builtin_nontemporal_load(&flat_netpin[s + i]);
          const float x = pos[idx];              // RT: keep pos L2-resident
          const float y = pos[num_pins + idx];
          xs[i] = x; ys[i] = y;
          mxx = fmaxf(mxx, x); mnx = fminf(mnx, x);
          mxy = fmaxf(mxy, y); mny = fminf(mny, y);
        }
      }
#pragma unroll
      for (int i = 0; i < 8; ++i) {
        if (i < deg) {
          spx += __expf((xs[i] - mxx) * inv_g);
          snx += __expf((mnx - xs[i]) * inv_g);
          spy += __expf((ys[i] - mxy) * inv_g);
          sny += __expf((mny - ys[i]) * inv_g);
        }
      }
    } else {
      // streaming two-pass fallback; second pass hits L2 (pos is L2-resident)
      for (int i = s; i < e; ++i) {
        const int idx = flat_netpin[i];
        const float x = pos[idx];
        const float y = pos[num_pins + idx];
        mxx = fmaxf(mxx, x); mnx = fminf(mnx, x);
        mxy = fmaxf(mxy, y); mny = fminf(mny, y);
      }
      for (int i = s; i < e; ++i) {
        const int idx = flat_netpin[i];
        const float x = pos[idx];
        const float y = pos[num_pins + idx];
        spx += __expf((x - mxx) * inv_g);
        snx += __expf((mnx - x) * inv_g);
        spy += __expf((y - mxy) * inv_g);
        sny += __expf((mny - y) * inv_g);
      }
    }

    acc += (mxx - mnx) + (mxy - mny)
         + g * (__logf(spx) + __logf(snx) + __logf(spy) + __logf(sny));
  }

  // fully converged here: EXEC all ones for the WMMA reduction
  const float b = block_sum(acc);
  if (threadIdx.x == 0) partials[blockIdx.x] = b;
}

__global__ void __launch_bounds__(WL_BLOCK)
wl_reduce_kernel(const float* __restrict__ partials, int n, float* __restrict__ out) {
  float acc = 0.0f;
  for (int i = threadIdx.x; i < n; i += WL_BLOCK) acc += partials[i];
  const float b = block_sum(acc);
  if (threadIdx.x == 0) out[0] = b;
}

extern "C" void kernel_launch(void* const* d_in, const int* in_sizes, int n_in,
                              void* d_out, int out_size, void* d_ws, size_t ws_size,
                              hipStream_t stream) {
  const float* pos          = (const float*)d_in[0];
  const int*   flat_netpin  = (const int*)d_in[1];
  const int*   netpin_start = (const int*)d_in[2];
  const int num_pins = in_sizes[1];
  const int num_nets = in_sizes[2] - 1;

  float* out      = (float*)d_out;
  float* partials = (float*)d_ws;

  int max_blocks = (int)(ws_size / sizeof(float));
  if (max_blocks > WL_MAX_BLOCKS) max_blocks = WL_MAX_BLOCKS;
  if (max_blocks < 1) max_blocks = 1;

  int nblocks = (num_nets + WL_BLOCK - 1) / WL_BLOCK;
  if (nblocks > max_blocks) nblocks = max_blocks;
  if (nblocks < 1) nblocks = 1;

  hipLaunchKernelGGL(wl_net_kernel, dim3(nblocks), dim3(WL_BLOCK), 0, stream,
                     pos, flat_netpin, netpin_start, partials, num_nets, num_pins);
  hipLaunchKernelGGL(wl_reduce_kernel, dim3(1), dim3(WL_BLOCK), 0, stream,
                     partials, nblocks, out);
}